// IrregularGRU_9663676416244
// MI455X (gfx1250) — compile-verified
//
#include <hip/hip_runtime.h>
#include <hip/hip_bf16.h>
#include <math.h>

#define B_   32
#define T_   4096
#define ND_  32
#define DH_  128
#define DZ_  128
#define G_   384      // 3*DZ
#define IN_  192      // 2*ND + DH

typedef float v2f __attribute__((ext_vector_type(2)));
typedef float v8f __attribute__((ext_vector_type(8)));

// D/C layout (32-bit 16x16): lane = {M%8 stripe}: M=(lane>>4)*8+v, N=lane&15
// A layout  (32-bit 16x4)  : M=lane&15, K=(lane>>4)*2 + vgpr
// B layout  (32-bit 4x16)  : N=lane&15, K=(lane>>4)*2 + vgpr

// ---------------------------------------------------------------------------
// Kernel 1: gi[rLocal][g] = x[b, k0+lt] . W_ih[g] + b_ih[g]
// rLocal = b*chunk + lt ; one wave per 16x16 output tile; K=192 in steps of 4.
// ---------------------------------------------------------------------------
__global__ __launch_bounds__(256)
void irgru_gi_gemm(const float* __restrict__ Y, const float* __restrict__ Mm,
                   const float* __restrict__ H, const float* __restrict__ Wih,
                   const float* __restrict__ bih, float* __restrict__ gi,
                   int k0, int chunkShift)
{
    const int chunk = 1 << chunkShift;
    const int wave  = threadIdx.x >> 5;
    const int lane  = threadIdx.x & 31;
    const int tile  = blockIdx.x * 8 + wave;
    const int rowTile = tile / 24;
    const int colTile = tile - rowTile * 24;

    const int m  = lane & 15;
    const int kl = (lane >> 4) << 1;              // 0 or 2
    const int rLocal = rowTile * 16 + m;
    const int b  = rLocal >> chunkShift;
    const int lt = rLocal & (chunk - 1);
    const int t  = k0 + lt;
    const int rowBase = b * T_ + t;

    const float* yp = Y   + rowBase * ND_;
    const float* mp = Mm  + rowBase * ND_;
    const float* hp = H   + rowBase * DH_;
    const int g     = colTile * 16 + m;
    const float* wp = Wih + g * IN_;

    v8f acc = {};
#pragma unroll
    for (int k = 0; k < 32; k += 4) {             // Y segment
        v2f a  = *(const v2f*)(yp + k + kl);
        v2f bb = *(const v2f*)(wp + k + kl);
        acc = __builtin_amdgcn_wmma_f32_16x16x4_f32(false, a, false, bb,
                                                    (short)0, acc, false, false);
    }
#pragma unroll
    for (int k = 32; k < 64; k += 4) {            // M segment
        v2f a  = *(const v2f*)(mp + (k - 32) + kl);
        v2f bb = *(const v2f*)(wp + k + kl);
        acc = __builtin_amdgcn_wmma_f32_16x16x4_f32(false, a, false, bb,
                                                    (short)0, acc, false, false);
    }
#pragma unroll
    for (int k = 64; k < 192; k += 4) {           // h segment
        v2f a  = *(const v2f*)(hp + (k - 64) + kl);
        v2f bb = *(const v2f*)(wp + k + kl);
        acc = __builtin_amdgcn_wmma_f32_16x16x4_f32(false, a, false, bb,
                                                    (short)0, acc, false, false);
    }

    const float bias   = bih[g];
    const int rowHalf  = rowTile * 16 + ((lane >> 4) << 3);
#pragma unroll
    for (int v = 0; v < 8; ++v)
        gi[(rowHalf + v) * G_ + g] = acc[v] + bias;
}

// ---------------------------------------------------------------------------
// Kernel 2: sequential GRU-ODE scan over one T-chunk. 2 blocks x 16 batches.
// Per step: z_d (elementwise, LDS) -> gh = z_d @ W_hh^T (f32 WMMA, W_hh
// B-fragments pinned in VGPRs) -> gates/update (elementwise) -> store Z_traj.
// ---------------------------------------------------------------------------
#define ZDS 130   // padded zd row stride (bank-conflict free, 8B aligned)
#define GHS 388   // padded gh row stride

__global__ __launch_bounds__(256)
void irgru_scan(const float* __restrict__ z0, const float* __restrict__ tdyn,
                const int* __restrict__ lens, const float* __restrict__ Whh,
                const float* __restrict__ bhh, const float* __restrict__ lg,
                const float* __restrict__ gi, float* __restrict__ wsz,
                float* __restrict__ out, int k0, int chunk)
{
    __shared__ float zd[16 * ZDS];
    __shared__ float gh[16 * GHS];
    __shared__ float gammaS[DZ_];

    const int tid  = threadIdx.x;
    const int bset = blockIdx.x;                 // 0 or 1 -> batches 16*bset..
    const int wave = tid >> 5;
    const int lane = tid & 31;
    const int m    = lane & 15;
    const int kl   = (lane >> 4) << 1;
    const int blh  = (lane >> 4) << 3;

    if (tid < DZ_) gammaS[tid] = log1pf(__expf(lg[tid]));   // softplus

    // recurrent state in registers: element e = j*256+tid -> (b=e>>7, d=e&127)
    float z[8];
#pragma unroll
    for (int j = 0; j < 8; ++j) {
        const int e = j * 256 + tid, bl = e >> 7, d = e & 127;
        const int bg = bset * 16 + bl;
        z[j] = (k0 == 0) ? z0[bg * DZ_ + d] : wsz[bg * DZ_ + d];
    }

    // pin this wave's W_hh B-fragments (3 N-tiles x 32 K-steps) in VGPRs
    v2f   wfrag[3][32];
    float wb[3];
#pragma unroll
    for (int c = 0; c < 3; ++c) {
        const int g = (wave + c * 8) * 16 + m;
        const float* wp = Whh + g * DZ_;
        wb[c] = bhh[g];
#pragma unroll
        for (int q = 0; q < 32; ++q)
            wfrag[c][q] = *(const v2f*)(wp + q * 4 + kl);
    }
    __syncthreads();

    for (int k = k0; k < k0 + chunk; ++k) {
        // phase 1: decay  z_d = z * exp(-gamma*dt)
#pragma unroll
        for (int j = 0; j < 8; ++j) {
            const int e = j * 256 + tid, bl = e >> 7, d = e & 127;
            const int bg = bset * 16 + bl;
            const float tk = tdyn[bg * T_ + k];
            const float tp = tdyn[bg * T_ + (k ? k - 1 : 0)];
            const float dt = fmaxf(tk - tp, 0.0f);
            zd[bl * ZDS + d] = z[j] * __expf(-gammaS[d] * dt);
        }
        __syncthreads();

        // phase 2: gh = z_d @ W_hh^T + b_hh   (M=16, N=384, K=128, f32 WMMA)
#pragma unroll
        for (int c = 0; c < 3; ++c) {
            v8f acc = {};
#pragma unroll
            for (int q = 0; q < 32; ++q) {
                v2f a = *(const v2f*)(&zd[m * ZDS + q * 4 + kl]);
                acc = __builtin_amdgcn_wmma_f32_16x16x4_f32(
                        false, a, false, wfrag[c][q], (short)0, acc, false, false);
            }
            const int g = (wave + c * 8) * 16 + m;
#pragma unroll
            for (int v = 0; v < 8; ++v)
                gh[(blh + v) * GHS + g] = acc[v] + wb[c];
        }
        __syncthreads();

        // phase 3: gates + state update + Z_traj store
#pragma unroll
        for (int j = 0; j < 8; ++j) {
            const int e = j * 256 + tid, bl = e >> 7, d = e & 127;
            const int bg = bset * 16 + bl;
            const int giRow = bg * chunk + (k - k0);
            const float* gir = gi + giRow * G_;
            const float i_r = gir[d];
            const float i_u = gir[d + DZ_];
            const float i_n = gir[d + 2 * DZ_];
            const float h_r = gh[bl * GHS + d];
            const float h_u = gh[bl * GHS + d + DZ_];
            const float h_n = gh[bl * GHS + d + 2 * DZ_];
            const float r  = 1.0f / (1.0f + __expf(-(i_r + h_r)));
            const float u  = 1.0f / (1.0f + __expf(-(i_u + h_u)));
            const float n  = tanhf(i_n + r * h_n);
            const float zdv = zd[bl * ZDS + d];
            const float zn  = (1.0f - u) * n + u * zdv;
            const float zo  = (k < lens[bg]) ? zn : zdv;
            z[j] = zo;
            out[(B_ * DZ_) + (bg * T_ + k) * DZ_ + d] = zo;   // Z_traj
            if (k + 1 < k0 + chunk) __builtin_prefetch(gir + G_, 0, 0);
        }
        __syncthreads();
    }

    // carry-out: z_final to d_out on last chunk, else to workspace
#pragma unroll
    for (int j = 0; j < 8; ++j) {
        const int e = j * 256 + tid, bl = e >> 7, d = e & 127;
        const int bg = bset * 16 + bl;
        if (k0 + chunk >= T_) out[bg * DZ_ + d] = z[j];
        else                  wsz[bg * DZ_ + d] = z[j];
    }
}

// ---------------------------------------------------------------------------
extern "C" void kernel_launch(void* const* d_in, const int* in_sizes, int n_in,
                              void* d_out, int out_size, void* d_ws, size_t ws_size,
                              hipStream_t stream) {
    (void)in_sizes; (void)n_in; (void)out_size;
    const float* z0   = (const float*)d_in[0];
    const float* tdyn = (const float*)d_in[1];
    const float* Y    = (const float*)d_in[2];
    const float* Mm   = (const float*)d_in[3];
    const int*   lens = (const int*)  d_in[4];
    const float* H    = (const float*)d_in[5];
    const float* Wih  = (const float*)d_in[6];
    const float* bih  = (const float*)d_in[7];
    const float* Whh  = (const float*)d_in[8];
    const float* bhh  = (const float*)d_in[9];
    const float* lg   = (const float*)d_in[10];
    float* out = (float*)d_out;
    float* wsz = (float*)d_ws;                   // B*DZ carry state
    float* gi  = wsz + B_ * DZ_;                 // chunked gi buffer

    // largest chunk (cap 1024 so gi <= ~50MB stays L2-resident) fitting ws
    int chunkShift = 10;
    while (chunkShift > 4) {
        size_t need = (size_t)B_ * DZ_ * 4 +
                      (size_t)B_ * ((size_t)1 << chunkShift) * G_ * 4;
        if (need <= ws_size) break;
        --chunkShift;
    }
    const int chunk = 1 << chunkShift;

    for (int k0 = 0; k0 < T_; k0 += chunk) {
        const int blocks = 6 * chunk;            // (32*chunk/16)*24 tiles / 8 waves
        irgru_gi_gemm<<<blocks, 256, 0, stream>>>(Y, Mm, H, Wih, bih, gi,
                                                  k0, chunkShift);
        irgru_scan<<<2, 256, 0, stream>>>(z0, tdyn, lens, Whh, bhh, lg,
                                          gi, wsz, out, k0, chunk);
    }
}